// QuanvolutionHybrid_65481071399824
// MI455X (gfx1250) — compile-verified
//
#include <hip/hip_runtime.h>
#include <math.h>

#define NBATCH   8192
#define NP       196        // num patches
#define NPAD     208        // 13 * 16
#define NT       13         // 16-wide tiles
#define FLATN    784
#define FEAT     980        // FLATN + NP
#define THRESH   0.8f
#define EPSF     1e-12f
#define BLOCK    416        // 13 waves: one wave per gram tile-column strip

typedef __attribute__((ext_vector_type(2))) float v2f;
typedef __attribute__((ext_vector_type(8))) float v8f;

__global__ __launch_bounds__(BLOCK) void quanv_hybrid_kernel(
    const float* __restrict__ x,       // (8192,1,28,28)
    const float* __restrict__ conv_w,  // (4,1,2,2)
    const float* __restrict__ conv_b,  // (4,)
    const float* __restrict__ lin_w,   // (10,980)
    const float* __restrict__ lin_b,   // (10,)
    float* __restrict__ out)           // (8192,10)
{
    __shared__ float s_feat[FEAT];       // [0,784): conv flat, [784,980): deg
    __shared__ float s_nrm[NPAD * 4];    // normalized pf rows, zero padded
    __shared__ float s_logits[10];

    const int tid  = threadIdx.x;
    const int lane = tid & 31;
    const int wave = tid >> 5;           // 0..12
    const int b    = blockIdx.x;
    const float* img = x + (size_t)b * 784;

    // ---- stage 1: 2x2/stride-2 conv (one thread per spatial position) ----
    if (tid < NP) {
        const int h = tid / 14, w = tid % 14;
        const float* p = img + (2 * h) * 28 + 2 * w;
        const float x00 = p[0], x01 = p[1], x10 = p[28], x11 = p[29];
#pragma unroll
        for (int c = 0; c < 4; ++c) {
            float v = conv_b[c]
                    + x00 * conv_w[c * 4 + 0] + x01 * conv_w[c * 4 + 1]
                    + x10 * conv_w[c * 4 + 2] + x11 * conv_w[c * 4 + 3];
            s_feat[c * NP + tid] = v;   // channel-major flatten (torch .view)
        }
    }
    __syncthreads();

    // ---- stage 2: pf row p = flat[4p..4p+3]; L2-normalize; zero padding ---
    if (tid < NPAD) {
        float f0 = 0.f, f1 = 0.f, f2 = 0.f, f3 = 0.f;
        if (tid < NP) {
            f0 = s_feat[4 * tid + 0]; f1 = s_feat[4 * tid + 1];
            f2 = s_feat[4 * tid + 2]; f3 = s_feat[4 * tid + 3];
            const float inv = 1.0f / (sqrtf(f0*f0 + f1*f1 + f2*f2 + f3*f3) + EPSF);
            f0 *= inv; f1 *= inv; f2 *= inv; f3 *= inv;
        }
        s_nrm[4 * tid + 0] = f0; s_nrm[4 * tid + 1] = f1;
        s_nrm[4 * tid + 2] = f2; s_nrm[4 * tid + 3] = f3;
    }
    __syncthreads();

    // ---- stage 3: gram via v_wmma_f32_16x16x4_f32, one tile-column per wave
    // A (16x4 f32) layout: lanes 0-15 hold K=0,1 ; lanes 16-31 hold K=2,3.
    // B (4x16 f32) gathers identically since B[k][n] = nrm[n][k].
    // deg[p] == column sum (w symmetric): column n lives entirely in lanes
    // n and n+16 -> integer pass-count per lane (v_cmp + carry-in add).
    // Diagonal removed once at the end (== reference's  w * (1 - eye)).
    {
        const int koff = (lane < 16) ? 0 : 2;
        const int l15  = lane & 15;
        const int n    = wave * 16 + l15;       // owned global column index

        const v2f bv = *(const v2f*)&s_nrm[n * 4 + koff];  // loop-invariant
        unsigned int cnt = 0u;
#pragma unroll
        for (int ti = 0; ti < NT; ++ti) {       // constant ti -> imm offsets
            const v2f av = *(const v2f*)&s_nrm[(ti * 16 + l15) * 4 + koff];
            v8f cv = {};
            cv = __builtin_amdgcn_wmma_f32_16x16x4_f32(
                     false, av, false, bv, (short)0, cv, false, false);
#pragma unroll
            for (int r = 0; r < 8; ++r) {
                const float fid = cv[r] * cv[r];        // squared cosine
                cnt += (fid >= THRESH);                 // cmp + carry-in add
            }
        }
        // self-fidelity of column n from the register-resident operand
        float dp = bv.x * bv.x + bv.y * bv.y;
        dp += __shfl_xor(dp, 16);                       // full 4-comp dot
        int total = (int)cnt + __shfl_xor((int)cnt, 16);// merge row-halves
        total -= (dp * dp >= THRESH) ? 1 : 0;           // drop diagonal edge
        if (lane < 16 && n < NP)
            s_feat[FLATN + n] = (float)total;           // exclusive owner
    }
    __syncthreads();

    // ---- stage 4: logits; wave j (<10) owns class j -----------------------
    if (wave < 10) {
        const int j = wave;
        float acc = 0.0f;
        for (int i = lane; i < FEAT; i += 32)           // single clean loop
            acc += s_feat[i] * lin_w[j * FEAT + i];
        acc += __shfl_xor(acc, 16);
        acc += __shfl_xor(acc, 8);
        acc += __shfl_xor(acc, 4);
        acc += __shfl_xor(acc, 2);
        acc += __shfl_xor(acc, 1);
        if (lane == 0) s_logits[j] = acc + lin_b[j];
    }
    __syncthreads();

    // ---- stage 5: log_softmax, write 10 outputs ---------------------------
    if (tid == 0) {
        float mx = s_logits[0];
#pragma unroll
        for (int j = 1; j < 10; ++j) mx = fmaxf(mx, s_logits[j]);
        float se = 0.0f;
#pragma unroll
        for (int j = 0; j < 10; ++j) se += expf(s_logits[j] - mx);
        const float lz = mx + logf(se);
#pragma unroll
        for (int j = 0; j < 10; ++j)
            out[(size_t)b * 10 + j] = s_logits[j] - lz;
    }
}

extern "C" void kernel_launch(void* const* d_in, const int* in_sizes, int n_in,
                              void* d_out, int out_size, void* d_ws, size_t ws_size,
                              hipStream_t stream) {
    const float* x      = (const float*)d_in[0];
    const float* conv_w = (const float*)d_in[1];
    const float* conv_b = (const float*)d_in[2];
    const float* lin_w  = (const float*)d_in[3];
    const float* lin_b  = (const float*)d_in[4];
    float* out = (float*)d_out;
    quanv_hybrid_kernel<<<NBATCH, BLOCK, 0, stream>>>(x, conv_w, conv_b,
                                                      lin_w, lin_b, out);
}